// MPNN_1417339207735
// MI455X (gfx1250) — compile-verified
//
#include <hip/hip_runtime.h>
#include <math.h>

// ---------------------------------------------------------------------------
// 3-layer GAT (PyG GATConv semantics, eval mode) for MI455X / gfx1250.
// fp32 WMMA (V_WMMA_F32_16X16X4_F32) for dense projections, W staged in LDS.
// Fout is a template constant so all A/B fragment loads and C stores are
// immediate-offset (global_load_b64 / ds_load_b32 / global_store_b32) from a
// single base register -> no address VALU in the 32x-unrolled WMMA loop.
// Edge phase is wave-parallel with L2-resident vectorized gathers + fp32
// atomics (node table = 51 MB < 192 MB L2).
// ---------------------------------------------------------------------------

typedef float v2f __attribute__((ext_vector_type(2)));
typedef float v8f __attribute__((ext_vector_type(8)));

#define DIN 128   // input feature width for every layer (128 = 2*64)
#define CCH 64    // channels per head

// ---------------- GEMM: C[N,FOUT] = A[N,128] @ B[128,FOUT] via f32 WMMA -----
// blockDim = 256 (8 waves). W staged into LDS once per block (64/32 KB),
// each wave computes one 16x16 tile, K=128 fully unrolled in steps of 4.
// FOUT=128: 8 col-tiles -> block covers 16 rows.
// FOUT=64 : 4 col-tiles -> 2 row-tiles per block -> block covers 32 rows.
template <int FOUT>
__global__ void gat_gemm_wmma(const float* __restrict__ A,
                              float* __restrict__ C,
                              const float* __restrict__ B, int N) {
  extern __shared__ float wlds[];          // DIN * FOUT floats

  // stage W into LDS, b128 granularity, fully coalesced
  constexpr int totalW = DIN * FOUT;       // 16384 or 8192 floats
#pragma unroll
  for (int i = 0; i < totalW / (256 * 4); ++i) {
    const int idx = (i * 256 + threadIdx.x) * 4;
    *(float4*)&wlds[idx] = *(const float4*)&B[idx];
  }
  __syncthreads();

  const int wave = threadIdx.x >> 5;
  const int lane = threadIdx.x & 31;
  constexpr int tilesX = FOUT >> 4;        // 8 or 4
  constexpr int rowTiles = 8 / tilesX;     // 1 or 2
  const int rt = wave / tilesX;
  const int ct = wave % tilesX;
  const int row0 = (blockIdx.x * rowTiles + rt) * 16;
  const int col0 = ct * 16;
  if (row0 >= N) return;                   // whole-wave exit keeps EXEC all-1s

  const int m  = row0 + (lane & 15);       // A fragment: lane -> row M
  const int n  = col0 + (lane & 15);       // B fragment: lane -> col N
  const int kb = (lane >> 4) * 2;          // half-wave selects K pair {0,1}/{2,3}

  const float* __restrict__ ap = A + (size_t)m * DIN + kb;   // A frag base
  const float* __restrict__ bp = &wlds[kb * FOUT + n];       // B frag base

  v8f acc = {};
#pragma unroll
  for (int k0 = 0; k0 < DIN; k0 += 4) {
    v2f a, b;
    a.x = ap[k0];                          // imm-offset global_load_b64
    a.y = ap[k0 + 1];
    b.x = bp[k0 * FOUT];                   // imm-offset ds_load_b32
    b.y = bp[(k0 + 1) * FOUT];             // imm-offset ds_load_b32
    acc = __builtin_amdgcn_wmma_f32_16x16x4_f32(
        /*neg_a=*/false, a, /*neg_b=*/false, b,
        /*c_mod=*/(short)0, acc, /*reuse_a=*/false, /*reuse_b=*/false);
  }

  // D layout: VGPR v holds M = (lane>=16 ? 8 : 0) + v, col = lane&15.
  const int rbase = row0 + (lane >> 4) * 8;
  float* __restrict__ cp = C + (size_t)rbase * FOUT + n;
#pragma unroll
  for (int v = 0; v < 8; ++v)
    cp[v * FOUT] = acc[v];                 // imm-offset global_store_b32
}

// ---------------- attention coefficients: es/ed[node*H+h] -------------------
// One wave per (node, head); lane-strided dot over C=64 channels, shfl reduce.
__global__ void gat_attn(const float* __restrict__ xl,
                         const float* __restrict__ as_,
                         const float* __restrict__ ad_,
                         float* __restrict__ es, float* __restrict__ ed,
                         int N, int H) {
  const int gw = (int)((blockIdx.x * blockDim.x + threadIdx.x) >> 5);
  const int lane = threadIdx.x & 31;
  if (gw >= N * H) return;
  const int node = gw / H;
  const int h = gw % H;
  const float* __restrict__ xp = xl + (size_t)node * H * CCH + h * CCH;
  const float* __restrict__ sp = as_ + h * CCH;
  const float* __restrict__ dp = ad_ + h * CCH;
  float s = 0.f, d = 0.f;
#pragma unroll
  for (int c = lane; c < CCH; c += 32) {
    float x = xp[c];
    s += x * sp[c];
    d += x * dp[c];
  }
#pragma unroll
  for (int off = 16; off > 0; off >>= 1) {
    s += __shfl_xor(s, off, 32);
    d += __shfl_xor(d, off, 32);
  }
  if (lane == 0) { es[gw] = s; ed[gw] = d; }
}

__global__ void fill_f32(float* __restrict__ p, float v, int n) {
  int i = blockIdx.x * blockDim.x + threadIdx.x;
  if (i < n) p[i] = v;
}

// float atomic max via signed/unsigned bit-pattern ordering trick
__device__ __forceinline__ void atomicMaxF(float* addr, float val) {
  if (val >= 0.f) atomicMax((int*)addr, __float_as_int(val));
  else            atomicMin((unsigned int*)addr, __float_as_uint(val));
}

// ---------------- edge pass 1: logits + segment max -------------------------
__global__ void edge_max_k(const int* __restrict__ ei,
                           const float* __restrict__ es,
                           const float* __restrict__ ed,
                           float* __restrict__ ebuf, float* __restrict__ mx,
                           int E, int N, int H) {
  const int i = blockIdx.x * blockDim.x + threadIdx.x;
  if (i >= E + N) return;
  int s, d;
  if (i < E) { s = ei[i]; d = ei[E + i]; } else { s = d = i - E; }
  for (int h = 0; h < H; ++h) {
    float e = es[s * H + h] + ed[d * H + h];
    e = (e >= 0.f) ? e : 0.2f * e;            // leaky_relu(0.2)
    ebuf[(size_t)i * H + h] = e;
    atomicMaxF(&mx[d * H + h], e);
  }
}

// ---------------- edge pass 2: exp + segment sum ----------------------------
__global__ void edge_exp_k(const int* __restrict__ ei,
                           float* __restrict__ ebuf,
                           const float* __restrict__ mx,
                           float* __restrict__ den,
                           int E, int N, int H) {
  const int i = blockIdx.x * blockDim.x + threadIdx.x;
  if (i >= E + N) return;
  const int d = (i < E) ? ei[E + i] : (i - E);
  for (int h = 0; h < H; ++h) {
    float ex = __expf(ebuf[(size_t)i * H + h] - mx[d * H + h]);
    ebuf[(size_t)i * H + h] = ex;
    atomicAdd(&den[d * H + h], ex);
  }
}

// ---------------- edge pass 3: normalized scatter-add -----------------------
// One wave per edge. Lane l owns channels [l*CH, l*CH+CH) with CH = FOUT/32,
// so the xl[src] gather is one b128/b64 load per lane and each lane's channels
// belong to a single head (64 % CH == 0). Scatter via fp32 atomics into the
// L2-resident accumulator.
template <int FOUT, int H>
__global__ void edge_aggr_k(const int* __restrict__ ei,
                            const float* __restrict__ ebuf,
                            const float* __restrict__ den,
                            const float* __restrict__ xl,
                            float* __restrict__ acc, int E, int N) {
  constexpr int CH = FOUT / 32;              // floats per lane (4 or 2)
  const int gw = (int)((blockIdx.x * blockDim.x + threadIdx.x) >> 5);
  const int lane = threadIdx.x & 31;
  if (gw >= E + N) return;
  // speculative prefetch of edge indices far ahead (global_prefetch_b8)
  if (gw + 8192 < E && lane == 0) {
    __builtin_prefetch(&ei[gw + 8192], 0, 0);
    __builtin_prefetch(&ei[E + gw + 8192], 0, 0);
  }
  int s, d;
  if (gw < E) { s = ei[gw]; d = ei[E + gw]; } else { s = d = gw - E; }

  const int h = (lane * CH) / CCH;           // head owning this lane's channels
  const float alpha =
      ebuf[(size_t)gw * H + h] / (den[d * H + h] + 1e-16f);
  const float* __restrict__ xp = &xl[(size_t)s * FOUT + lane * CH];
  float* __restrict__ ap = &acc[(size_t)d * FOUT + lane * CH];
  float xv[CH];
#pragma unroll
  for (int j = 0; j < CH; ++j) xv[j] = xp[j];  // merged -> b128/b64 load
#pragma unroll
  for (int j = 0; j < CH; ++j) atomicAdd(ap + j, xv[j] * alpha);
}

// ---------------- epilogue: bias + BatchNorm(eval) [+ ReLU] -----------------
__global__ void gat_finalize(const float* __restrict__ acc,
                             const float* __restrict__ b,
                             const float* __restrict__ g,
                             const float* __restrict__ be,
                             const float* __restrict__ m,
                             const float* __restrict__ v,
                             float* __restrict__ out,
                             int total, int F, int doRelu) {
  const int i = blockIdx.x * blockDim.x + threadIdx.x;
  if (i >= total) return;
  const int c = i & (F - 1);                 // F is 128 or 64 (pow2)
  float z = acc[i] + b[c];
  z = (z - m[c]) * rsqrtf(v[c] + 1e-5f) * g[c] + be[c];
  if (doRelu) z = fmaxf(z, 0.f);
  out[i] = z;
}

// ---------------------------------------------------------------------------
template <int FOUT, int H>
static void launch_layer(const float* hin, const float* W, const float* as_,
                         const float* ad_, const float* b, const float* g,
                         const float* be, const float* m, const float* v,
                         int N, int E, const int* ei,
                         float* xl, float* acc, float* es, float* ed,
                         float* mx, float* den, float* ebuf,
                         float* out, int doRelu, hipStream_t stream) {
  const int ET = E + N;
  // GEMM (dynamic LDS = DIN*FOUT floats: 64 KB or 32 KB; WGP has 320 KB)
  constexpr int rowsPerBlock = 16 * (8 / (FOUT >> 4));
  const size_t ldsBytes = (size_t)DIN * FOUT * sizeof(float);
  gat_gemm_wmma<FOUT><<<(N + rowsPerBlock - 1) / rowsPerBlock, 256, ldsBytes,
                        stream>>>(hin, xl, W, N);
  // attention scalars
  gat_attn<<<(N * H + 7) / 8, 256, 0, stream>>>(xl, as_, ad_, es, ed, N, H);
  // init segment buffers
  hipMemsetAsync(den, 0, (size_t)N * H * sizeof(float), stream);
  hipMemsetAsync(acc, 0, (size_t)N * FOUT * sizeof(float), stream);
  fill_f32<<<(N * H + 255) / 256, 256, 0, stream>>>(mx, -INFINITY, N * H);
  // segment softmax + aggregation
  const int eb = (ET + 255) / 256;
  edge_max_k<<<eb, 256, 0, stream>>>(ei, es, ed, ebuf, mx, E, N, H);
  edge_exp_k<<<eb, 256, 0, stream>>>(ei, ebuf, mx, den, E, N, H);
  edge_aggr_k<FOUT, H><<<(ET + 7) / 8, 256, 0, stream>>>(ei, ebuf, den, xl,
                                                         acc, E, N);
  // epilogue
  const int tot = N * FOUT;
  gat_finalize<<<(tot + 255) / 256, 256, 0, stream>>>(acc, b, g, be, m, v, out,
                                                      tot, FOUT, doRelu);
}

extern "C" void kernel_launch(void* const* d_in, const int* in_sizes, int n_in,
                              void* d_out, int out_size, void* d_ws, size_t ws_size,
                              hipStream_t stream) {
  const float* x   = (const float*)d_in[0];
  const int*   ei  = (const int*)d_in[1];
  const float* w1  = (const float*)d_in[2];
  const float* as1 = (const float*)d_in[3];
  const float* ad1 = (const float*)d_in[4];
  const float* b1  = (const float*)d_in[5];
  const float* w2  = (const float*)d_in[6];
  const float* as2 = (const float*)d_in[7];
  const float* ad2 = (const float*)d_in[8];
  const float* b2  = (const float*)d_in[9];
  const float* w3  = (const float*)d_in[10];
  const float* as3 = (const float*)d_in[11];
  const float* ad3 = (const float*)d_in[12];
  const float* b3  = (const float*)d_in[13];
  const float* g1  = (const float*)d_in[14];
  const float* be1 = (const float*)d_in[15];
  const float* m1  = (const float*)d_in[16];
  const float* v1  = (const float*)d_in[17];
  const float* g2  = (const float*)d_in[18];
  const float* be2 = (const float*)d_in[19];
  const float* m2  = (const float*)d_in[20];
  const float* v2  = (const float*)d_in[21];
  const float* g3  = (const float*)d_in[22];
  const float* be3 = (const float*)d_in[23];
  const float* m3  = (const float*)d_in[24];
  const float* v3  = (const float*)d_in[25];

  const int N  = in_sizes[0] / DIN;   // 100000
  const int E  = in_sizes[1] / 2;     // 1600000
  const int ET = E + N;

  // workspace layout (floats): xl, acc, z, es, ed, mx, den, ebuf  (~172 MB)
  float* ws = (float*)d_ws;
  size_t off = 0;
  float* xl   = ws + off; off += (size_t)N * DIN;
  float* acc  = ws + off; off += (size_t)N * DIN;
  float* zb   = ws + off; off += (size_t)N * DIN;
  float* es   = ws + off; off += (size_t)N * 2;
  float* ed   = ws + off; off += (size_t)N * 2;
  float* mx   = ws + off; off += (size_t)N * 2;
  float* den  = ws + off; off += (size_t)N * 2;
  float* ebuf = ws + off; off += (size_t)ET * 2;
  (void)off; (void)ws_size; (void)n_in;

  // layer 1: in=x[N,128], H=2, Fout=128, BN+ReLU -> zb
  launch_layer<128, 2>(x,  w1, as1, ad1, b1, g1, be1, m1, v1, N, E, ei,
                       xl, acc, es, ed, mx, den, ebuf, zb, 1, stream);
  // layer 2: in=zb, H=2, Fout=128, BN+ReLU -> zb (gemm reads zb first;
  // finalize overwrites zb from acc afterwards)
  launch_layer<128, 2>(zb, w2, as2, ad2, b2, g2, be2, m2, v2, N, E, ei,
                       xl, acc, es, ed, mx, den, ebuf, zb, 1, stream);
  // layer 3: in=zb, H=1, Fout=64, mean over 1 head = identity, BN, no ReLU -> d_out
  launch_layer<64, 1>(zb, w3, as3, ad3, b3, g3, be3, m3, v3, N, E, ei,
                      xl, acc, es, ed, mx, den, ebuf, (float*)d_out, 0, stream);
}